// DynamicUpsamplingFilter_3C_48000554500521
// MI455X (gfx1250) — compile-verified
//
#include <hip/hip_runtime.h>
#include <cstdint>
#include <cstddef>

// DynamicUpsamplingFilter: out[b,c,h,w] = sum_{dy,dx} xpad[b,c,h+dy-1,w+dx-1] * f[b,dy*3+dx,h,w]
// Bandwidth-bound (AI ~2.2 flop/B). CDNA5 path: TDM async global->LDS staging of x,
// double-buffered on TENSORcnt, fp32 VALU FMAs, NT streaming stores.

typedef unsigned int u32x4 __attribute__((ext_vector_type(4)));
typedef int          i32x8 __attribute__((ext_vector_type(8)));
typedef int          i32x4 __attribute__((ext_vector_type(4)));

namespace {
constexpr int kB = 4, kC = 128, kH = 180, kW = 320, kF = 9;
constexpr int WT      = 64;        // output pixels (w) per workgroup
constexpr int CBK     = 16;        // channels per TDM chunk
constexpr int NCHUNK  = kC / CBK;  // 8
constexpr int NTHREADS = 256;      // 8 waves (wave32)
}

// Low 32 bits of a generic pointer to LDS == byte offset within LDS.
__device__ __forceinline__ unsigned ldsOffset(const void* p) {
  return (unsigned)(uintptr_t)p;
}

// One 2-D TDM transfer: tile_dim0 = WT floats along w (stride 1 element),
// tile_dim1 = CBK channels (stride = H*W elements). Lands contiguously in LDS
// as [CBK][WT] floats starting at ldsByteOff.
__device__ __forceinline__ void tdmLoadPlane(const float* gsrc, unsigned ldsByteOff) {
  uint64_t ga = (uint64_t)(uintptr_t)gsrc;
  u32x4 g0;
  g0.x = 0x1u;                                               // count=1, user D#, no gather
  g0.y = ldsByteOff;                                         // lds_addr (bytes)
  g0.z = (unsigned)ga;                                       // global_addr[31:0]
  g0.w = (unsigned)((ga >> 32) & 0x1FFFFFFull) | (2u << 30); // addr[56:32] | type=2
  i32x8 g1;
  g1[0] = 2 << 16;                 // data_size = 2 (4 bytes); no multicast/pad/iterate
  g1[1] = (WT  & 0xFFFF) << 16;    // tensor_dim0[15:0]  (== tile: no OOB use)
  g1[2] = (CBK & 0xFFFF) << 16;    // tensor_dim0[31:16]=0 | tensor_dim1[15:0]
  g1[3] = (WT  & 0xFFFF) << 16;    // tensor_dim1[31:16]=0 | tile_dim0
  g1[4] = (CBK & 0xFFFF);          // tile_dim1 | tile_dim2=0 (2-D)
  g1[5] = kH * kW;                 // tensor_dim0_stride[31:0] = channel stride
  g1[6] = 0;                       // stride0[47:32] | stride1[15:0]
  g1[7] = 0;                       // stride1[47:16]
  i32x4 gz4 = {0, 0, 0, 0};        // groups 2/3 unused (tile_dim2=0)
  i32x8 gz8 = {0, 0, 0, 0, 0, 0, 0, 0};
  // 6-arg form (clang-23 / therock-10.0 headers)
  __builtin_amdgcn_tensor_load_to_lds(g0, g1, gz4, gz4, gz8, 0);
}

__global__ __launch_bounds__(NTHREADS)
void duf_tdm_kernel(const float* __restrict__ x,
                    const float* __restrict__ f,
                    float* __restrict__ out) {
  __shared__ float xbuf[2][3][CBK][WT];  // double-buffered [row][chan][w], 24 KB
  __shared__ float halo[2][3][CBK];      // [side(l/r)][row][chan]
  __shared__ float flds[kF][WT];         // per-pixel filter taps

  const int wt  = blockIdx.x * WT;
  const int h   = blockIdx.y;
  const int b   = blockIdx.z;
  const int tid = threadIdx.x;
  const int wl  = tid & (WT - 1);  // 0..63
  const int cz  = tid >> 6;        // 0..3

  const bool hasTop = (h > 0);
  const bool hasBot = (h < kH - 1);
  const bool interior = hasTop && hasBot;

  // ---- stage filters into LDS (coalesced) ----
  for (int idx = tid; idx < kF * WT; idx += NTHREADS) {
    int t = idx >> 6, w2 = idx & (WT - 1);
    flds[t][w2] = f[(((size_t)b * kF + t) * kH + h) * kW + wt + w2];
  }

  // ---- pre-zero halo rows in both buffers when at an h edge (TDM skips them) ----
  if (!hasTop) {
    for (int idx = tid; idx < 2 * CBK * WT; idx += NTHREADS) {
      int bi = idx >= CBK * WT;
      (&xbuf[bi][0][0][0])[idx - bi * CBK * WT] = 0.0f;
    }
  }
  if (!hasBot) {
    for (int idx = tid; idx < 2 * CBK * WT; idx += NTHREADS) {
      int bi = idx >= CBK * WT;
      (&xbuf[bi][2][0][0])[idx - bi * CBK * WT] = 0.0f;
    }
  }

  // ---- kick off TDM for chunk 0 (wave 0 only; TDM ignores EXEC) ----
  if (tid < 32) {
    const size_t cbase0 = ((size_t)b * kC + 0) * (size_t)(kH * kW);
#pragma unroll
    for (int r = 0; r < 3; ++r) {
      int hh = h - 1 + r;
      if (hh >= 0 && hh < kH)
        tdmLoadPlane(x + cbase0 + (size_t)hh * kW + wt, ldsOffset(&xbuf[0][r][0][0]));
    }
  }

  __syncthreads();  // filters + zeroed rows visible

  // per-thread filter taps live in registers for the whole kernel
  float fr[kF];
#pragma unroll
  for (int t = 0; t < kF; ++t) fr[t] = flds[t][wl];

  for (int i = 0; i < NCHUNK; ++i) {
    const int bi = i & 1;

    // issue TDM for next chunk into the other buffer (overlaps this chunk's compute)
    if (tid < 32 && (i + 1) < NCHUNK) {
      const size_t cbase = ((size_t)b * kC + (size_t)(i + 1) * CBK) * (size_t)(kH * kW);
#pragma unroll
      for (int r = 0; r < 3; ++r) {
        int hh = h - 1 + r;
        if (hh >= 0 && hh < kH)
          tdmLoadPlane(x + cbase + (size_t)hh * kW + wt, ldsOffset(&xbuf[(i + 1) & 1][r][0][0]));
      }
    }

    // halo columns (w = wt-1 and w = wt+WT) for this chunk: 2*3*16 = 96 values
    if (tid < 96) {
      int side = tid & 1;
      int r    = (tid >> 1) % 3;
      int c    = tid / 6;
      int hh   = h - 1 + r;
      int wc   = side ? (wt + WT) : (wt - 1);
      float v  = 0.0f;
      if (hh >= 0 && hh < kH && wc >= 0 && wc < kW)
        v = x[((size_t)(b * kC + i * CBK + c) * kH + hh) * kW + wc];
      halo[side][r][c] = v;
    }

    // wave 0: make sure THIS chunk's DMA has landed (next chunk's may stay in flight)
    if (tid < 32) {
      if ((i + 1) < NCHUNK) {
        if (interior) __builtin_amdgcn_s_wait_tensorcnt(3);
        else          __builtin_amdgcn_s_wait_tensorcnt(2);
      } else {
        __builtin_amdgcn_s_wait_tensorcnt(0);
      }
    }
    __syncthreads();  // DMA + halo visible to all waves

    // compute: 4 channels per thread, 9 FMAs each
#pragma unroll
    for (int cc4 = 0; cc4 < 4; ++cc4) {
      const int c = cz * 4 + cc4;
      float acc = 0.0f;
#pragma unroll
      for (int r = 0; r < 3; ++r) {
        const float* row = &xbuf[bi][r][c][0];
        float xl = (wl == 0)      ? halo[0][r][c] : row[wl - 1];
        float xc = row[wl];
        float xr = (wl == WT - 1) ? halo[1][r][c] : row[wl + 1];
        acc = fmaf(xl, fr[r * 3 + 0], acc);
        acc = fmaf(xc, fr[r * 3 + 1], acc);
        acc = fmaf(xr, fr[r * 3 + 2], acc);
      }
      __builtin_nontemporal_store(
          acc, &out[((size_t)(b * kC + i * CBK + c) * kH + h) * kW + wt + wl]);
    }

    __syncthreads();  // everyone done reading xbuf[bi] before it is DMA'd over
  }
}

extern "C" void kernel_launch(void* const* d_in, const int* in_sizes, int n_in,
                              void* d_out, int out_size, void* d_ws, size_t ws_size,
                              hipStream_t stream) {
  (void)in_sizes; (void)n_in; (void)out_size; (void)d_ws; (void)ws_size;
  const float* x = (const float*)d_in[0];
  const float* f = (const float*)d_in[1];
  float* out = (float*)d_out;
  dim3 grid(kW / WT, kH, kB);  // (5, 180, 4)
  duf_tdm_kernel<<<grid, NTHREADS, 0, stream>>>(x, f, out);
}